// CMCAccuracy_11441792876991
// MI455X (gfx1250) — compile-verified
//
#include <hip/hip_runtime.h>

// CMC accuracy for MI455X (gfx1250, wave32).
// Strategy: fp32 WMMA (v_wmma_f32_16x16x4_f32) GEMM for -2*e@e.T,
// exact streaming top-5 selection instead of the reference's full argsort.
// score(i,j) = norms[j] - 2*dot(i,j)   (sq[i] shift does not change ranking).

#define N_TOTAL 16384
#define DIM     256
#define GSIZE   8192          // rows per flank
#define TM      16            // rows per workgroup
#define WAVES   8
#define THREADS 256
#define COLS_PER_WAVE (GSIZE / WAVES)        // 1024
#define TILES_PER_WAVE (COLS_PER_WAVE / 16)  // 64
#define KSTEPS  (DIM / 4)                    // 64 wmma k-steps of 4
#define TOPK    5
#define BIGF    3.402823466e+38f

typedef float v2f __attribute__((ext_vector_type(2)));
typedef float v8f __attribute__((ext_vector_type(8)));

__global__ void cmc_init_kernel(int* __restrict__ count) {
    *count = 0;
}

// One wave32 per row: norms[r] = sum_k E[r][k]^2
__global__ __launch_bounds__(256)
void cmc_norms_kernel(const float* __restrict__ E, float* __restrict__ norms) {
    int wave = threadIdx.x >> 5;
    int lane = threadIdx.x & 31;
    int row  = blockIdx.x * 8 + wave;
    const float* p = E + (size_t)row * DIM;
    float s = 0.0f;
#pragma unroll
    for (int t = 0; t < DIM / 32; ++t) {
        float v = p[lane + 32 * t];
        s = fmaf(v, v, s);
    }
#pragma unroll
    for (int off = 16; off > 0; off >>= 1)
        s += __shfl_xor(s, off, 32);
    if (lane == 0) norms[row] = s;
}

__device__ __forceinline__ void top5_insert(float s, int mt, float td[TOPK], int tm[TOPK]) {
    if (s < td[TOPK - 1]) {
        td[TOPK - 1] = s; tm[TOPK - 1] = mt;
#pragma unroll
        for (int u = TOPK - 1; u > 0; --u) {
            if (td[u] < td[u - 1]) {
                float x = td[u]; td[u] = td[u - 1]; td[u - 1] = x;
                int   y = tm[u]; tm[u] = tm[u - 1]; tm[u - 1] = y;
            }
        }
    }
}

__global__ __launch_bounds__(THREADS)
void cmc_main_kernel(const float* __restrict__ E,
                     const int*   __restrict__ labels,
                     const float* __restrict__ norms,
                     int*         __restrict__ count) {
    __shared__ float tileBuf[WAVES][TM][17];      // padded: avoid bank conflicts
    __shared__ float mergeD[WAVES][TM][TOPK];
    __shared__ int   mergeM[WAVES][TM][TOPK];
    __shared__ int   ldsLabels[GSIZE];            // 32 KB: this flank's labels

    const int tid  = threadIdx.x;
    const int wave = tid >> 5;
    const int lane = tid & 31;
    const int hi   = lane >> 4;                   // K-half selector (ISA 16x4 f32 layout)
    const int ln   = lane & 15;

    const int rowBase = blockIdx.x * TM;          // global first row of this block
    const int flank   = rowBase / GSIZE;
    const int colBase = flank * GSIZE;

    // Stage flank labels into LDS (coalesced).
    for (int i = tid; i < GSIZE; i += THREADS)
        ldsLabels[i] = labels[colBase + i];
    __syncthreads();

    // ---- A fragments: 16x256 row panel, held entirely in registers ----
    // lane L (L&15 = M) holds {E[M][4t+2*hi], E[M][4t+2*hi+1]} per k-step t.
    const int   arow = rowBase + ln;
    const float* Ap  = E + (size_t)arow * DIM + 2 * hi;
    v2f afrag[KSTEPS];
#pragma unroll
    for (int t = 0; t < KSTEPS; ++t)
        afrag[t] = *(const v2f*)(Ap + 4 * t);

    // ---- per-lane selection state: one row, 8 of 16 tile-columns ----
    const int myRow   = ln;
    const int myLabel = ldsLabels[rowBase - colBase + myRow];
    const int csBase  = hi * 8;
    float td[TOPK]; int tm[TOPK];
#pragma unroll
    for (int u = 0; u < TOPK; ++u) { td[u] = BIGF; tm[u] = 0; }

    int jcol0 = colBase + wave * COLS_PER_WAVE;
    for (int t = 0; t < TILES_PER_WAVE; ++t, jcol0 += 16) {
        // B fragment rows are embedding rows j (same layout/pattern as A).
        const float* Bp = E + (size_t)(jcol0 + ln) * DIM + 2 * hi;
        v8f c = {};
#pragma unroll
        for (int kk = 0; kk < KSTEPS; ++kk) {
            v2f b = *(const v2f*)(Bp + 4 * kk);
            c = __builtin_amdgcn_wmma_f32_16x16x4_f32(
                    false, afrag[kk], false, b, (short)0, c, false, false);
        }
        // Epilogue: score = norms[j] - 2*dot.  c[r] is row (r+8*hi), col ln.
        const float nb = norms[jcol0 + ln];
#pragma unroll
        for (int r = 0; r < 8; ++r)
            tileBuf[wave][r + 8 * hi][ln] = fmaf(-2.0f, c[r], nb);
        __syncthreads();

        // Selection scan: lane owns row myRow, tile-cols csBase..csBase+7.
#pragma unroll
        for (int q = 0; q < 8; ++q) {
            const int c0 = csBase + q;
            const int j  = jcol0 + c0;                // global column index
            float s  = tileBuf[wave][myRow][c0];
            int   mt = (ldsLabels[j - colBase] == myLabel) ? 1 : 0;
            if (j != rowBase + myRow)                 // exclude self
                top5_insert(s, mt, td, tm);
        }
        __syncthreads();
    }

    // Merge the two half-lane partial top-5s (lane L gets lane L+16's list).
#pragma unroll
    for (int u = 0; u < TOPK; ++u) {
        float od = __shfl(td[u], lane + 16, 32);
        int   om = __shfl(tm[u], lane + 16, 32);
        if (hi == 0) top5_insert(od, om, td, tm);
    }
    if (hi == 0) {
#pragma unroll
        for (int u = 0; u < TOPK; ++u) {
            mergeD[wave][ln][u] = td[u];
            mergeM[wave][ln][u] = tm[u];
        }
    }
    __syncthreads();

    // Wave 0: final 8-way merge per row, then count matched rows.
    bool matched = false;
    if (tid < TM) {
        float fd[TOPK]; int fm[TOPK];
#pragma unroll
        for (int u = 0; u < TOPK; ++u) { fd[u] = BIGF; fm[u] = 0; }
        for (int w = 0; w < WAVES; ++w)
#pragma unroll
            for (int u = 0; u < TOPK; ++u)
                top5_insert(mergeD[w][tid][u], mergeM[w][tid][u], fd, fm);
#pragma unroll
        for (int u = 0; u < TOPK; ++u) matched |= (fm[u] != 0);
    }
    if (tid < 32) {
        unsigned long long mask = __ballot(matched);
        int cnt = __popcll(mask & 0xFFFFull);
        if (tid == 0) atomicAdd(count, cnt);
    }
}

__global__ void cmc_finalize_kernel(const int* __restrict__ count,
                                    float* __restrict__ out) {
    // mean over flanks of per-flank means == global mean (equal flank sizes)
    out[0] = (float)(*count) / (float)N_TOTAL;
}

extern "C" void kernel_launch(void* const* d_in, const int* in_sizes, int n_in,
                              void* d_out, int out_size, void* d_ws, size_t ws_size,
                              hipStream_t stream) {
    (void)in_sizes; (void)n_in; (void)out_size; (void)ws_size;
    const float* E      = (const float*)d_in[0];
    const int*   labels = (const int*)d_in[1];
    // d_in[2] (flanks) unused: contiguous equal-sized groups == reshape.
    // d_in[3] (k) hardcoded to 5 (fixed-size register top-k).

    float* norms = (float*)d_ws;                                   // 16384 f32
    int*   count = (int*)((char*)d_ws + N_TOTAL * sizeof(float));  // 1 int

    hipLaunchKernelGGL(cmc_init_kernel, dim3(1), dim3(1), 0, stream, count);
    hipLaunchKernelGGL(cmc_norms_kernel, dim3(N_TOTAL / 8), dim3(256), 0, stream,
                       E, norms);
    hipLaunchKernelGGL(cmc_main_kernel, dim3(N_TOTAL / TM), dim3(THREADS), 0, stream,
                       E, labels, norms, count);
    hipLaunchKernelGGL(cmc_finalize_kernel, dim3(1), dim3(1), 0, stream,
                       count, (float*)d_out);
}